// SoftArgsortTopK_10007273799760
// MI455X (gfx1250) — compile-verified
//
#include <hip/hip_runtime.h>
#include <hip/hip_bf16.h>
#include <math.h>

// Problem constants (fixed by setup_inputs in the reference)
#define BB 16
#define RR 1024
#define TT 16
#define UU 16
#define DD 128
#define KK 16            // top-k
#define INV_REG 100.0f   // 1/REG_STRENGTH
#define INV_TEMP 10.0f   // 1/TEMPERATURE

typedef __attribute__((ext_vector_type(2))) float v2f;
typedef __attribute__((ext_vector_type(8))) float v8f;

// ---------------------------------------------------------------------------
// Kernel 1: per-batch ascending bitonic sort (LDS) + sequential PAV isotonic
// regression (non-increasing) + write yk[b][0..15] to d_out.
//   z_i = a_i + (R-i)/reg  (a = ascending-sorted scores)
//   sol = PAV_nonincreasing(z);  out_i = sol_i - (R-i)/reg;  yk = out[R-16..R-1]
// One block of 1024 threads per batch (32 wave32s).
// ---------------------------------------------------------------------------
__global__ __launch_bounds__(RR)
void sort_pav_kernel(const float* __restrict__ scores, float* __restrict__ yk_out) {
    __shared__ float a[RR];      // sorted scores
    __shared__ float bmean[RR];  // PAV block means
    __shared__ int   bcnt[RR];   // PAV block counts
    __shared__ float sol[RR];    // expanded isotonic solution

    const int b   = blockIdx.x;
    const int tid = threadIdx.x;

    a[tid] = scores[b * RR + tid];

    // Bitonic sort, ascending
    for (unsigned size = 2; size <= RR; size <<= 1) {
        for (unsigned stride = size >> 1; stride > 0; stride >>= 1) {
            __syncthreads();
            unsigned i = tid;
            unsigned j = i ^ stride;
            if (j > i) {
                float x = a[i], y = a[j];
                bool up = ((i & size) == 0);
                bool sw = up ? (x > y) : (x < y);
                if (sw) { a[i] = y; a[j] = x; }
            }
        }
    }
    __syncthreads();

    // Sequential PAV by thread 0 (tiny work; merges are ~never triggered
    // because w steps by 100 >> score spread, but implement exactly).
    if (tid == 0) {
        int top = -1;
        for (int i = 0; i < RR; ++i) {
            float z = a[i] + (float)(RR - i) * INV_REG;
            ++top; bmean[top] = z; bcnt[top] = 1;
            while (top > 0 && bmean[top] > bmean[top - 1]) {
                float c0 = (float)bcnt[top - 1], c1 = (float)bcnt[top];
                bmean[top - 1] = (bmean[top - 1] * c0 + bmean[top] * c1) / (c0 + c1);
                bcnt[top - 1] += bcnt[top];
                --top;
            }
        }
        int pos = 0;
        for (int blk = 0; blk <= top; ++blk) {
            float mval = bmean[blk];
            for (int c = 0; c < bcnt[blk]; ++c) sol[pos++] = mval;
        }
    }
    __syncthreads();

    if (tid < KK) {
        int i = RR - KK + tid;
        yk_out[b * KK + tid] = sol[i] - (float)(RR - i) * INV_REG;
    }
}

// ---------------------------------------------------------------------------
// Kernel 2: soft_mask[b,kk,r] = softmax_r(-|scores[b,r]-yk[b,kk]|/T) -> d_ws
//           soft_label_expectation[b,kk] = sum_r mask * ref_class -> d_out
// One block of 256 threads per (b,kk); 4 elements of r per thread in regs.
// ---------------------------------------------------------------------------
__global__ __launch_bounds__(256)
void softmax_mask_kernel(const float* __restrict__ scores,
                         const int*   __restrict__ ref_class,
                         const float* __restrict__ yk,
                         float* __restrict__ mask_ws,
                         float* __restrict__ lab_out) {
    __shared__ float redm[256];
    __shared__ float reds[256];
    __shared__ float redl[256];

    const int b   = blockIdx.x >> 4;
    const int kk  = blockIdx.x & 15;
    const int tid = threadIdx.x;
    const float y = yk[b * KK + kk];

    float v[4];
    float cls[4];
    for (int j = 0; j < 4; ++j) {
        int r = tid + j * 256;
        float s = scores[b * RR + r];
        v[j]   = -fabsf(s - y) * INV_TEMP;
        cls[j] = (float)ref_class[b * RR + r];
    }

    // max reduction
    float mx = fmaxf(fmaxf(v[0], v[1]), fmaxf(v[2], v[3]));
    redm[tid] = mx;
    __syncthreads();
    for (int s = 128; s > 0; s >>= 1) {
        if (tid < s) redm[tid] = fmaxf(redm[tid], redm[tid + s]);
        __syncthreads();
    }
    float gmax = redm[0];

    float e[4];
    float psum = 0.0f, plab = 0.0f;
    for (int j = 0; j < 4; ++j) {
        e[j] = expf(v[j] - gmax);
        psum += e[j];
        plab += e[j] * cls[j];
    }
    reds[tid] = psum;
    redl[tid] = plab;
    __syncthreads();
    for (int s = 128; s > 0; s >>= 1) {
        if (tid < s) { reds[tid] += reds[tid + s]; redl[tid] += redl[tid + s]; }
        __syncthreads();
    }
    float inv = 1.0f / reds[0];

    float* mrow = mask_ws + (size_t)(b * KK + kk) * RR;
    for (int j = 0; j < 4; ++j) {
        int r = tid + j * 256;
        mrow[r] = e[j] * inv;
    }
    if (tid == 0) lab_out[b * KK + kk] = redl[0] * inv;
}

// ---------------------------------------------------------------------------
// Kernel 3: batched GEMMs via V_WMMA_F32_16X16X4_F32 (exact f32 math).
//   C_dp[b]  = mask[b](16x1024) @ DP[b](1024x256)
//   C_psi[b] = mask[b](16x1024) @ psi[b](1024x2048)
// One wave32 per (b, 32-column slab): 2 N-tiles (16 cols each) per wave so
// the A fragment load is amortized over 2 WMMAs. 1152 waves total.
//
// A frag (16x4 f32): lane m=lane&15, half=lane>>4:
//   a = { A[m][kb+2*half], A[m][kb+2*half+1] }          (one b64 load)
// B frag (4x16 f32):  n=lane&15:
//   bx = B[kb+2*half][n], by = B[kb+2*half+1][n]        (two b32 loads)
// C/D (16x16 f32): c[g] = C[g + 8*half][n]
// ---------------------------------------------------------------------------
__global__ __launch_bounds__(32)
void wmma_gemm_kernel(const float* __restrict__ mask,
                      const float* __restrict__ DP,
                      const float* __restrict__ psi,
                      float* __restrict__ out_dp,
                      float* __restrict__ out_psi) {
    const int TILES_DP  = (TT * UU) / 32;  // 8 slabs of 32 cols
    const int TILES_PSI = (TT * DD) / 32;  // 64 slabs
    const int PER_B     = TILES_DP + TILES_PSI;  // 72

    const int wid = blockIdx.x;
    const int b   = wid / PER_B;
    const int t   = wid % PER_B;

    const float* Bmat;
    float*       out;
    int ldB, col0;
    if (t < TILES_DP) {
        Bmat = DP + (size_t)b * RR * (TT * UU);
        out  = out_dp + (size_t)b * KK * (TT * UU);
        ldB  = TT * UU;
        col0 = t * 32;
    } else {
        int tt = t - TILES_DP;
        Bmat = psi + (size_t)b * RR * (TT * DD);
        out  = out_psi + (size_t)b * KK * (TT * DD);
        ldB  = TT * DD;
        col0 = tt * 32;
    }

    const int lane = threadIdx.x;
    const int m    = lane & 15;   // row of A / col of B
    const int half = lane >> 4;

    const float* Arow = mask + (size_t)(b * KK + m) * RR + 2 * half;
    const float* Bp   = Bmat + (size_t)(2 * half) * ldB + col0 + m;

    v8f acc0 = {};
    v8f acc1 = {};

    #pragma unroll 4
    for (int kb = 0; kb < RR; kb += 4) {
        v2f afrag = *(const v2f*)(Arow + kb);
        const float* bp = Bp + (size_t)kb * ldB;
        v2f b0, b1;
        b0.x = bp[0];
        b0.y = bp[ldB];
        b1.x = bp[16];
        b1.y = bp[16 + ldB];
        // D = A*B + C ; (neg_a, A, neg_b, B, c_mod, C, reuse_a, reuse_b)
        acc0 = __builtin_amdgcn_wmma_f32_16x16x4_f32(
            false, afrag, false, b0, (short)0, acc0, false, false);
        acc1 = __builtin_amdgcn_wmma_f32_16x16x4_f32(
            false, afrag, false, b1, (short)0, acc1, false, false);
    }

    #pragma unroll
    for (int g = 0; g < 8; ++g) {
        int row = g + 8 * half;
        out[(size_t)row * ldB + col0 + m]      = acc0[g];
        out[(size_t)row * ldB + col0 + 16 + m] = acc1[g];
    }
}

// ---------------------------------------------------------------------------
extern "C" void kernel_launch(void* const* d_in, const int* in_sizes, int n_in,
                              void* d_out, int out_size, void* d_ws, size_t ws_size,
                              hipStream_t stream) {
    (void)in_sizes; (void)n_in; (void)out_size; (void)ws_size;

    const float* scores    = (const float*)d_in[0];
    const int*   ref_class = (const int*)d_in[1];
    const float* DP        = (const float*)d_in[2];
    const float* psi       = (const float*)d_in[3];
    // d_in[4] = k (16), d_in[5] = K (8, unused by reference) — shapes fixed.

    float* out      = (float*)d_out;
    float* yk       = out;                          // [16,16]
    float* lab      = out + BB * KK;                // [16,16]
    float* out_dp   = out + 2 * BB * KK;            // [16,16,16,16]
    float* out_psi  = out_dp + BB * KK * TT * UU;   // [16,16,16,128]

    float* mask_ws  = (float*)d_ws;                 // [16,16,1024] f32 = 1 MB

    sort_pav_kernel<<<BB, RR, 0, stream>>>(scores, yk);
    softmax_mask_kernel<<<BB * KK, 256, 0, stream>>>(scores, ref_class, yk,
                                                     mask_ws, lab);
    const int PER_B = (TT * UU) / 32 + (TT * DD) / 32;  // 72
    wmma_gemm_kernel<<<BB * PER_B, 32, 0, stream>>>(mask_ws, DP, psi,
                                                    out_dp, out_psi);
}